// HybridMoELayer_38543036514486
// MI455X (gfx1250) — compile-verified
//
#include <hip/hip_runtime.h>

#define H 1024
#define NE 8
#define NTOK 8192
#define MOE_EPS 1e-10f

#define BM 128
#define BN 128
#define BK 32

typedef float v2f __attribute__((ext_vector_type(2)));
typedef float v4f __attribute__((ext_vector_type(4)));
typedef float v8f __attribute__((ext_vector_type(8)));

#if defined(__AMDGCN__)
typedef __attribute__((address_space(3))) float lds_f;
// Async DMA: 16 bytes per lane, global -> LDS, tracked by ASYNCcnt.
#define ASYNC_LD16(l, g)                                             \
  asm volatile("global_load_async_to_lds_b128 %0, %1, off" ::"v"(l), \
               "v"(g)                                                \
               : "memory")
#define WAIT_ASYNC() asm volatile("s_wait_asynccnt 0x0" ::: "memory")
#else
typedef float lds_f;
#define ASYNC_LD16(l, g) ((void)(l), (void)(g))
#define WAIT_ASYNC() ((void)0)
#endif

// ---------------------------------------------------------------------------
// Zero output + per-expert counters (stream-ordered before router).
// ---------------------------------------------------------------------------
__global__ void moe_zero_kernel(float* __restrict__ out,
                                int* __restrict__ counts, int total) {
  int stride = gridDim.x * blockDim.x;
  for (int i = blockIdx.x * blockDim.x + threadIdx.x; i < total; i += stride)
    out[i] = 0.0f;
  if (blockIdx.x == 0 && threadIdx.x < NE) counts[threadIdx.x] = 0;
}

// ---------------------------------------------------------------------------
// Router: one wave32 per token. 8 dot products (K=1024) reduced across the
// wave with __shfl_xor (width 32 on gfx1250), softmax + top-2 on lane 0,
// combine scale folded into the routing weight, bucket append via atomics.
// ---------------------------------------------------------------------------
__global__ __launch_bounds__(256)
void moe_router_kernel(const float* __restrict__ tokens,
                       const float* __restrict__ rw,
                       const float* __restrict__ rb,
                       int* __restrict__ counts,
                       int* __restrict__ btok,
                       float* __restrict__ bwgt) {
  const int lane = threadIdx.x & 31;
  const int wave = threadIdx.x >> 5;
  const int n = blockIdx.x * 8 + wave;
  if (n >= NTOK) return;

  const float* x = tokens + (size_t)n * H;
  float acc[NE];
#pragma unroll
  for (int e = 0; e < NE; ++e) acc[e] = 0.0f;

  for (int h = lane; h < H; h += 32) {
    const float xv = x[h];
#pragma unroll
    for (int e = 0; e < NE; ++e) acc[e] = fmaf(xv, rw[e * H + h], acc[e]);
  }
#pragma unroll
  for (int off = 16; off >= 1; off >>= 1) {
#pragma unroll
    for (int e = 0; e < NE; ++e) acc[e] += __shfl_xor(acc[e], off, 32);
  }

  if (lane == 0) {
    float lg[NE], p[NE];
    float mx = -3.4e38f;
#pragma unroll
    for (int e = 0; e < NE; ++e) {
      lg[e] = acc[e] + rb[e];
      mx = fmaxf(mx, lg[e]);
    }
    float s = 0.0f;
#pragma unroll
    for (int e = 0; e < NE; ++e) {
      p[e] = __expf(lg[e] - mx);
      s += p[e];
    }
    const float inv = 1.0f / s;
#pragma unroll
    for (int e = 0; e < NE; ++e) p[e] *= inv;

    int i0 = 0;
    float p0 = p[0];
#pragma unroll
    for (int e = 1; e < NE; ++e)
      if (p[e] > p0) { p0 = p[e]; i0 = e; }
    int i1 = (i0 == 0) ? 1 : 0;
    float p1 = p[i1];
#pragma unroll
    for (int e = 0; e < NE; ++e)
      if (e != i0 && p[e] > p1) { p1 = p[e]; i1 = e; }

    const float sw = p0 + p1;
    const float w0 = p0 / (sw + MOE_EPS);
    const float w1 = p1 / (sw + MOE_EPS);
    const float wsum = w0 + w1;
    // out = (sum_k w_k y_k) / clamp(wsum, eps) * (1 + wsum*eps)
    const float scale = (1.0f + wsum * MOE_EPS) / fmaxf(wsum, MOE_EPS);

    const int pos0 = atomicAdd(&counts[i0], 1);
    btok[i0 * NTOK + pos0] = n;
    bwgt[i0 * NTOK + pos0] = w0 * scale;
    const int pos1 = atomicAdd(&counts[i1], 1);
    btok[i1 * NTOK + pos1] = n;
    bwgt[i1 * NTOK + pos1] = w1 * scale;
  }
}

// ---------------------------------------------------------------------------
// Grouped expert GEMM, fp32 WMMA (V_WMMA_F32_16X16X4_F32), double-buffered
// LDS staged via GLOBAL_LOAD_ASYNC_TO_LDS_B128 (ASYNCcnt), so next K-chunk
// DMA overlaps WMMA on the current chunk. A tile uses a 16B-chunk XOR swizzle
// (chunk ^= row&7) to spread fragment reads across LDS banks while keeping
// every async copy 16B-aligned.
// ---------------------------------------------------------------------------
__global__ __launch_bounds__(256, 2)
void moe_expert_gemm_kernel(const float* __restrict__ tokens,
                            const float* __restrict__ ew,
                            const int* __restrict__ counts,
                            const int* __restrict__ btok,
                            const float* __restrict__ bwgt,
                            float* __restrict__ out) {
  const int e = blockIdx.y;
  const int cnt = counts[e];
  const int m0 = blockIdx.x * BM;
  if (m0 >= cnt) return;
  const int n0 = blockIdx.z * BN;

  __shared__ float As[2][BM][BK];  // XOR-swizzled 16B chunks
  __shared__ float Bs[2][BK][BN];
  __shared__ int row_tok[BM];
  __shared__ float row_w[BM];

  const int tid = threadIdx.x;

  // Gather row metadata (token id + pre-scaled routing weight).
  for (int r = tid; r < BM; r += 256) {
    const int gm = m0 + r;
    if (gm < cnt) {
      row_tok[r] = btok[e * NTOK + gm];
      row_w[r] = bwgt[e * NTOK + gm];
    } else {
      row_tok[r] = 0;
      row_w[r] = 0.0f;
    }
  }
  __syncthreads();

  const int lane = tid & 31;
  const int wv = tid >> 5;
  const int wm = wv & 1;        // 0..1 -> 64 rows each
  const int wn = wv >> 1;       // 0..3 -> 32 cols each
  const int lhalf = lane >> 4;  // 0: K{0,1}/M=v ; 1: K{2,3}/M=v+8
  const int lrow = lane & 15;

  const v8f vzero = {0.f, 0.f, 0.f, 0.f, 0.f, 0.f, 0.f, 0.f};
  v8f acc[4][2];
#pragma unroll
  for (int i = 0; i < 4; ++i)
#pragma unroll
    for (int j = 0; j < 2; ++j) acc[i][j] = vzero;

  const float* ewe = ew + (size_t)e * H * H;

  // A staging: 2 threads/row, 4x16B chunks each. B: 8 threads/k-row.
  const int ar = tid >> 1;
  const int ac4 = (tid & 1) * 4;  // first 16B-chunk index in row (0 or 4)
  const float* arow = tokens + (size_t)row_tok[ar] * H;
  const int bkr = tid >> 3;
  const int bc = (tid & 7) * 16;

  auto stage = [&](int buf, int k0) {
    const float* ga = arow + k0 + ac4 * 4;
#pragma unroll
    for (int q = 0; q < 4; ++q) {
      const int c4s = (ac4 + q) ^ (ar & 7);  // swizzled chunk
      ASYNC_LD16((lds_f*)&As[buf][ar][c4s << 2], ga + (q << 2));
    }
    const float* gb = ewe + (size_t)(k0 + bkr) * H + n0 + bc;
#pragma unroll
    for (int q = 0; q < 4; ++q) {
      ASYNC_LD16((lds_f*)&Bs[buf][bkr][bc + (q << 2)], gb + (q << 2));
    }
  };

  // Prologue: stage chunk 0, wait, barrier.
  stage(0, 0);
  WAIT_ASYNC();
  __syncthreads();

  for (int k0 = 0, it = 0; k0 < H; k0 += BK, ++it) {
    const int buf = it & 1;
    if (k0 + BK < H) stage(buf ^ 1, k0 + BK);  // DMA next chunk during compute

#pragma unroll
    for (int kk = 0; kk < BK; kk += 4) {
      const int ka = kk + 2 * lhalf;
      // Swizzled A column: uniform per lane (m&7 == lrow&7 for all i).
      const int colA = (((ka >> 2) ^ (lrow & 7)) << 2) | (ka & 3);
      v2f a[4], b[2];
#pragma unroll
      for (int i = 0; i < 4; ++i) {
        const int m = wm * 64 + i * 16 + lrow;
        a[i].x = As[buf][m][colA];
        a[i].y = As[buf][m][colA + 1];
      }
#pragma unroll
      for (int j = 0; j < 2; ++j) {
        const int nn = wn * 32 + j * 16 + lrow;
        b[j].x = Bs[buf][ka][nn];
        b[j].y = Bs[buf][ka + 1][nn];
      }
#pragma unroll
      for (int i = 0; i < 4; ++i)
#pragma unroll
        for (int j = 0; j < 2; ++j)
          acc[i][j] = __builtin_amdgcn_wmma_f32_16x16x4_f32(
              false, a[i], false, b[j], (short)0, acc[i][j], false, false);
    }

    WAIT_ASYNC();     // next-chunk DMA landed
    __syncthreads();  // all waves done reading buf; safe to overwrite next it
  }

  // Weighted scatter-accumulate: C/D layout -> lane<16: M=v, lane>=16: M=v+8.
#pragma unroll
  for (int i = 0; i < 4; ++i) {
#pragma unroll
    for (int v = 0; v < 8; ++v) {
      const int r = wm * 64 + i * 16 + v + 8 * lhalf;
      if (m0 + r < cnt) {
        const int t = row_tok[r];
        const float wgt = row_w[r];
        float* orow = out + (size_t)t * H + n0;
#pragma unroll
        for (int j = 0; j < 2; ++j) {
          const int nn = wn * 32 + j * 16 + lrow;
          atomicAdd(orow + nn, wgt * acc[i][j][v]);
        }
      }
    }
  }
}

// ---------------------------------------------------------------------------
extern "C" void kernel_launch(void* const* d_in, const int* in_sizes, int n_in,
                              void* d_out, int out_size, void* d_ws,
                              size_t ws_size, hipStream_t stream) {
  const float* tokens = (const float*)d_in[0];  // [4,2048,1024] f32
  const float* rw     = (const float*)d_in[1];  // [8,1024]      f32
  const float* rb     = (const float*)d_in[2];  // [8]           f32
  const float* ew     = (const float*)d_in[3];  // [8,1024,1024] f32
  float* out = (float*)d_out;                   // [4,2048,1024] f32

  // Workspace layout: counts (8 ints, padded) | bucket tokens | bucket weights
  char* ws = (char*)d_ws;
  int* counts = (int*)ws;
  int* btok   = (int*)(ws + 256);
  float* bwgt = (float*)(ws + 256 + (size_t)NE * NTOK * sizeof(int));

  moe_zero_kernel<<<2048, 256, 0, stream>>>(out, counts, NTOK * H);
  moe_router_kernel<<<NTOK / 8, 256, 0, stream>>>(tokens, rw, rb, counts, btok,
                                                  bwgt);
  dim3 grid(NTOK / BM, NE, H / BN);  // worst-case M coverage; blocks early-exit
  moe_expert_gemm_kernel<<<grid, 256, 0, stream>>>(tokens, ew, counts, btok,
                                                   bwgt, out);
}